// SPDNetConfigurable_22643067584695
// MI455X (gfx1250) — compile-verified
//
#include <hip/hip_runtime.h>
#include <math.h>

// SPDNet forward for MI455X (gfx1250, wave32).
// One wave (32 lanes) per 32x32 SPD matrix:
//   S = W X W^T          -> 2 GEMMs via V_WMMA_F32_16X16X4_F32
//   eigh(S)              -> parallel cyclic Jacobi in LDS (16 disjoint pairs/round)
//   L = V diag(log l) V^T -> 1 GEMM via WMMA (scale folded into B fragments)
//   triu-vec * sqrt2      -> fused into 528->10 classifier with shuffle reduce.

typedef float v2f __attribute__((ext_vector_type(2)));
typedef float v8f __attribute__((ext_vector_type(8)));

#define LSTR 33        // LDS row stride (pad to kill bank conflicts)
#define NSWEEPS 8
#define THRESH 1e-4f

__device__ __forceinline__ v8f wmma4(v2f a, v2f b, v8f c) {
  // D = A(16x4,f32) * B(4x16,f32) + C(16x16,f32)
  return __builtin_amdgcn_wmma_f32_16x16x4_f32(
      /*neg_a=*/false, a, /*neg_b=*/false, b,
      /*c_mod=*/(short)0, c, /*reuse_a=*/false, /*reuse_b=*/false);
}

// ---- fragment loaders (layouts per CDNA5 ISA §7.12.2) ----
// A 16x4 f32: lanes 0-15 rows M, VGPR0/1 = K,K+1 ; lanes 16-31 = K+2,K+3
__device__ __forceinline__ v2f frag_a_lds(const float* M, int mr, int kc, int lane) {
  int row = mr + (lane & 15);
  int k0  = kc + ((lane >> 4) << 1);
  v2f a;
  a.x = M[row * LSTR + k0];
  a.y = M[row * LSTR + k0 + 1];
  return a;
}
__device__ __forceinline__ v2f frag_a_g32(const float* M, int mr, int kc, int lane) {
  int row = mr + (lane & 15);
  int k0  = kc + ((lane >> 4) << 1);
  v2f a;
  a.x = M[row * 32 + k0];
  a.y = M[row * 32 + k0 + 1];
  return a;
}
// B 4x16 f32: lanes 0-15 cols N, VGPR r holds K = kb + half*2 + r
__device__ __forceinline__ v2f frag_b_lds(const float* M, int kb, int nb, int lane) {
  int n  = nb + (lane & 15);
  int k0 = kb + ((lane >> 4) << 1);
  v2f b;
  b.x = M[k0 * LSTR + n];
  b.y = M[(k0 + 1) * LSTR + n];
  return b;
}
// B fragment of W^T read from row-major W in global memory
__device__ __forceinline__ v2f frag_bt_g32(const float* W, int kb, int nb, int lane) {
  int n  = nb + (lane & 15);
  int k0 = kb + ((lane >> 4) << 1);
  v2f b;
  b.x = W[n * 32 + k0];
  b.y = W[n * 32 + k0 + 1];
  return b;
}
// B fragment of diag(f) * V^T read from row-major V in LDS
__device__ __forceinline__ v2f frag_bt_lds_scaled(const float* V, const float* f,
                                                  int kb, int nb, int lane) {
  int n  = nb + (lane & 15);
  int k0 = kb + ((lane >> 4) << 1);
  v2f b;
  b.x = f[k0]     * V[n * LSTR + k0];
  b.y = f[k0 + 1] * V[n * LSTR + k0 + 1];
  return b;
}
// C/D 16x16 f32: VGPR r -> rows r (lanes 0-15) / r+8 (lanes 16-31)
__device__ __forceinline__ void store_c_lds(float* M, int mr, int nb, int lane, v8f c) {
  int col = nb + (lane & 15);
  int rb  = mr + ((lane >> 4) << 3);
#pragma unroll
  for (int r = 0; r < 8; ++r) M[(rb + r) * LSTR + col] = c[r];
}

// round-robin tournament pairing: 31 rounds x 16 disjoint pairs cover all 496 pairs
__device__ __forceinline__ void pairpq(int r, int j, int& p, int& q) {
  if (j == 0) { p = 31; q = r; }
  else        { p = (r + j) % 31; q = (r - j + 31) % 31; }
}

__global__ __launch_bounds__(32)
void spdnet_fused_kernel(const float* __restrict__ X, const float* __restrict__ W,
                         const float* __restrict__ Wc, const float* __restrict__ bias,
                         float* __restrict__ out) {
  __shared__ float Ab[32 * LSTR];   // working matrix (X -> S -> L)
  __shared__ float Vb[32 * LSTR];   // T = W X, then eigenvectors V
  __shared__ float cs[32];          // c/s per rotation pair
  __shared__ float fl[32];          // log(max(lambda, t))

  const int lane = threadIdx.x;
  const int m = blockIdx.x;
  const float* Xm = X + (size_t)m * 1024;

  // warm the cacheline path for this matrix (global_prefetch_b8)
  __builtin_prefetch(Xm + lane * 32, 0, 1);

  // ---- stage X into LDS (b128 loads) ----
  const float4* X4 = (const float4*)Xm;
#pragma unroll
  for (int i = 0; i < 8; ++i) {
    int f4 = i * 32 + lane;
    float4 v = X4[f4];
    int e = f4 * 4;
    int row = e >> 5, col = e & 31;
    Ab[row * LSTR + col + 0] = v.x;
    Ab[row * LSTR + col + 1] = v.y;
    Ab[row * LSTR + col + 2] = v.z;
    Ab[row * LSTR + col + 3] = v.w;
  }
  __syncthreads();

  // ---- GEMM1: T = W @ X  -> Vb ----
#pragma unroll
  for (int ti = 0; ti < 2; ++ti)
#pragma unroll
    for (int tj = 0; tj < 2; ++tj) {
      v8f acc;
#pragma unroll
      for (int e = 0; e < 8; ++e) acc[e] = 0.0f;
#pragma unroll
      for (int kc = 0; kc < 8; ++kc)
        acc = wmma4(frag_a_g32(W, ti * 16, kc * 4, lane),
                    frag_b_lds(Ab, kc * 4, tj * 16, lane), acc);
      store_c_lds(Vb, ti * 16, tj * 16, lane, acc);
    }
  __syncthreads();

  // ---- GEMM2: S = T @ W^T -> Ab ----
#pragma unroll
  for (int ti = 0; ti < 2; ++ti)
#pragma unroll
    for (int tj = 0; tj < 2; ++tj) {
      v8f acc;
#pragma unroll
      for (int e = 0; e < 8; ++e) acc[e] = 0.0f;
#pragma unroll
      for (int kc = 0; kc < 8; ++kc)
        acc = wmma4(frag_a_lds(Vb, ti * 16, kc * 4, lane),
                    frag_bt_g32(W, kc * 4, tj * 16, lane), acc);
      store_c_lds(Ab, ti * 16, tj * 16, lane, acc);
    }
  __syncthreads();

  // ---- symmetrize S (lane = row) ----
  {
    float tmp[32];
#pragma unroll
    for (int j = 0; j < 32; ++j)
      tmp[j] = 0.5f * (Ab[lane * LSTR + j] + Ab[j * LSTR + lane]);
    __syncthreads();
#pragma unroll
    for (int j = 0; j < 32; ++j) Ab[lane * LSTR + j] = tmp[j];
  }

  // ---- V = I ----
#pragma unroll
  for (int i = 0; i < 32; ++i) Vb[i * LSTR + lane] = (i == lane) ? 1.0f : 0.0f;
  __syncthreads();

  // ---- parallel cyclic Jacobi: A' = J^T A J,  V' = V J ----
  for (int sweep = 0; sweep < NSWEEPS; ++sweep) {
    for (int r = 0; r < 31; ++r) {
      if (lane < 16) {
        int p, q; pairpq(r, lane, p, q);
        float app = Ab[p * LSTR + p];
        float aqq = Ab[q * LSTR + q];
        float apq = Ab[p * LSTR + q];
        float c = 1.0f, s = 0.0f;
        if (fabsf(apq) > 1e-12f) {
          float tau = (aqq - app) / (2.0f * apq);
          float t = (tau >= 0.0f ? 1.0f : -1.0f) /
                    (fabsf(tau) + sqrtf(1.0f + tau * tau));
          c = 1.0f / sqrtf(1.0f + t * t);
          s = t * c;
        }
        cs[2 * lane]     = c;
        cs[2 * lane + 1] = s;
      }
      __syncthreads();

      // column rotations on A and V (lane = row)
#pragma unroll 4
      for (int j = 0; j < 16; ++j) {
        int p, q; pairpq(r, j, p, q);
        float c = cs[2 * j], s = cs[2 * j + 1];
        float ap = Ab[lane * LSTR + p], aq = Ab[lane * LSTR + q];
        Ab[lane * LSTR + p] = c * ap - s * aq;
        Ab[lane * LSTR + q] = s * ap + c * aq;
        float vp = Vb[lane * LSTR + p], vq = Vb[lane * LSTR + q];
        Vb[lane * LSTR + p] = c * vp - s * vq;
        Vb[lane * LSTR + q] = s * vp + c * vq;
      }
      __syncthreads();

      // row rotations on A (lane = column)
#pragma unroll 4
      for (int j = 0; j < 16; ++j) {
        int p, q; pairpq(r, j, p, q);
        float c = cs[2 * j], s = cs[2 * j + 1];
        float ap = Ab[p * LSTR + lane], aq = Ab[q * LSTR + lane];
        Ab[p * LSTR + lane] = c * ap - s * aq;
        Ab[q * LSTR + lane] = s * ap + c * aq;
      }
      __syncthreads();
    }
  }

  // ---- eigenvalues -> log(max(l, t)) ----
  fl[lane] = logf(fmaxf(Ab[lane * LSTR + lane], THRESH));
  __syncthreads();

  // ---- GEMM3: L = V @ (diag(fl) V^T) -> Ab ----
#pragma unroll
  for (int ti = 0; ti < 2; ++ti)
#pragma unroll
    for (int tj = 0; tj < 2; ++tj) {
      v8f acc;
#pragma unroll
      for (int e = 0; e < 8; ++e) acc[e] = 0.0f;
#pragma unroll
      for (int kc = 0; kc < 8; ++kc)
        acc = wmma4(frag_a_lds(Vb, ti * 16, kc * 4, lane),
                    frag_bt_lds_scaled(Vb, fl, kc * 4, tj * 16, lane), acc);
      store_c_lds(Ab, ti * 16, tj * 16, lane, acc);
    }
  __syncthreads();

  // ---- triu-vec (sqrt2 off-diag) fused with 528->10 classifier ----
  float acc[10];
#pragma unroll
  for (int o = 0; o < 10; ++o) acc[o] = 0.0f;
  const float SQRT2 = 1.41421356237309515f;
  for (int i = 0; i < 32; ++i) {
    if (i <= lane) {  // lane = column j, upper triangle i <= j
      float lv = Ab[i * LSTR + lane] * ((i == lane) ? 1.0f : SQRT2);
      int fidx = i * 32 - (i * (i - 1)) / 2 + (lane - i);
#pragma unroll
      for (int o = 0; o < 10; ++o) acc[o] += lv * Wc[o * 528 + fidx];
    }
  }
#pragma unroll
  for (int o = 0; o < 10; ++o) {
    float rsum = acc[o];
    for (int off = 16; off > 0; off >>= 1)
      rsum += __shfl_down(rsum, off, 32);
    if (lane == 0) out[m * 10 + o] = rsum + bias[o];
  }
}

extern "C" void kernel_launch(void* const* d_in, const int* in_sizes, int n_in,
                              void* d_out, int out_size, void* d_ws, size_t ws_size,
                              hipStream_t stream) {
  const float* X    = (const float*)d_in[0];
  const float* W    = (const float*)d_in[1];
  const float* Wc   = (const float*)d_in[2];
  const float* bias = (const float*)d_in[3];
  float* out = (float*)d_out;
  int B = in_sizes[0] / (32 * 32);   // 32768
  hipLaunchKernelGGL(spdnet_fused_kernel, dim3(B), dim3(32), 0, stream,
                     X, W, Wc, bias, out);
}